// O2P2Model_45157286150537
// MI455X (gfx1250) — compile-verified
//
#include <hip/hip_runtime.h>

// ---------------- problem constants ----------------
constexpr int kHW = 128 * 128;   // 16384
constexpr int kD  = 256;
constexpr int kPH = 1024;
constexpr int kRH = 1024;
constexpr int kTH = 512;
constexpr int kN  = 16;          // num objects

typedef float v2f __attribute__((ext_vector_type(2)));
typedef float v8f __attribute__((ext_vector_type(8)));

// D = A(16x4 f32) * B(4x16 f32) + C(16x16 f32)
__device__ __forceinline__ v8f wmma4(v2f a, v2f b, v8f c) {
  return __builtin_amdgcn_wmma_f32_16x16x4_f32(
      /*neg_a=*/false, a, /*neg_b=*/false, b,
      /*c_mod=*/(short)0, c, /*reuse_a=*/false, /*reuse_b=*/false);
}

__device__ __forceinline__ float sigmoidf_(float x) {
  return 1.0f / (1.0f + __expf(-x));
}

// ============================================================
// Perception GEMM1 (split-K): Ppart[kc][16][1024] partial of
//   pin[16,65536] @ Wp1[65536,1024]
// pin[o][k] = k < 3*HW ? img[k] : mask[o][k-3*HW]
// ============================================================
__global__ void k_percept1(const float* __restrict__ img,
                           const float* __restrict__ mask,
                           const float* __restrict__ Wp1,
                           float* __restrict__ Ppart) {
  const int nt   = blockIdx.x;       // 0..63 column tile
  const int kc   = blockIdx.y;       // 0..15 K chunk (4096 each)
  const int lane = threadIdx.x & 31;
  const int kh   = lane >> 4;        // wave half (K offset for A/B)
  const int l16  = lane & 15;
  const int n0   = nt * 16;
  const int kbeg = kc * 4096;
  const int kend = kbeg + 4096;

  v8f acc = {};
  const float* pB = Wp1 + (size_t)(kbeg + kh * 2) * kPH + n0 + l16;
  for (int k0 = kbeg; k0 < kend; k0 += 4) {
    const int ks = k0 + kh * 2;                      // 8B-aligned
    v2f a;
    if (ks < 3 * kHW) a = *(const v2f*)(img + ks);   // shared image rows
    else              a = *(const v2f*)(mask + l16 * kHW + (ks - 3 * kHW));
    v2f b;
    b.x = pB[0];
    b.y = pB[kPH];
    __builtin_prefetch((const void*)(pB + (size_t)16 * kPH), 0, 0);
    acc = wmma4(a, b, acc);
    pB += (size_t)4 * kPH;
  }
  float* out = Ppart + (size_t)kc * (16 * kPH);
#pragma unroll
  for (int r = 0; r < 8; ++r)
    out[(r + kh * 8) * kPH + n0 + l16] = acc[r];
}

// deterministic split-K reduce + bias + relu -> C1[16,1024]
__global__ void k_reduce1(const float* __restrict__ Ppart,
                          const float* __restrict__ bp1,
                          float* __restrict__ C1) {
  const int idx = blockIdx.x * blockDim.x + threadIdx.x;   // 0..16383
  const int n = idx & (kPH - 1);
  float s = bp1[n];
#pragma unroll
  for (int kc = 0; kc < 16; ++kc) s += Ppart[kc * 16 * kPH + idx];
  C1[idx] = fmaxf(s, 0.0f);
}

// ============================================================
// Generic M=16 GEMM: C[16,N] = act(A[16,K] @ B[K,N] + bias)
// one wave per 16-wide column tile (blockIdx.x)
// ============================================================
__global__ void k_gemm16(const float* __restrict__ A, int lda,
                         const float* __restrict__ B, int ldb,
                         const float* __restrict__ bias,
                         float* __restrict__ Cdst, int ldc,
                         int K, int relu) {
  const int lane = threadIdx.x & 31;
  const int kh   = lane >> 4;
  const int l16  = lane & 15;
  const int n0   = blockIdx.x * 16;

  v8f acc = {};
  const float* pA = A + l16 * lda + kh * 2;
  const float* pB = B + (size_t)(kh * 2) * ldb + n0 + l16;
  for (int k0 = 0; k0 < K; k0 += 4) {
    v2f a = *(const v2f*)pA;
    v2f b;
    b.x = pB[0];
    b.y = pB[ldb];
    acc = wmma4(a, b, acc);
    pA += 4;
    pB += (size_t)4 * ldb;
  }
  const float bv = bias[n0 + l16];
#pragma unroll
  for (int r = 0; r < 8; ++r) {
    float v = acc[r] + bv;
    if (relu) v = fmaxf(v, 0.0f);
    Cdst[(r + kh * 8) * ldc + n0 + l16] = v;
  }
}

// ============================================================
// Interaction GEMM1: ih[256,512] = relu(pair[256,512] @ Wi1 + bi1)
// pair[t*16+j][k] = k<256 ? obj[t][k] : obj[j][k-256]
// ============================================================
__global__ void k_inter1(const float* __restrict__ obj,
                         const float* __restrict__ Wi1,
                         const float* __restrict__ bi1,
                         float* __restrict__ ih) {
  const int nt   = blockIdx.x;   // 0..31
  const int t    = blockIdx.y;   // 0..15  (i index)
  const int lane = threadIdx.x & 31;
  const int kh   = lane >> 4;
  const int l16  = lane & 15;    // j index (tile row)
  const int n0   = nt * 16;

  v8f acc = {};
  const float* pB = Wi1 + (size_t)(kh * 2) * kTH + n0 + l16;
  for (int k0 = 0; k0 < 2 * kD; k0 += 4) {
    const int ks = k0 + kh * 2;
    v2f a;
    if (ks < kD) a = *(const v2f*)(obj + t * kD + ks);
    else         a = *(const v2f*)(obj + l16 * kD + (ks - kD));
    v2f b;
    b.x = pB[0];
    b.y = pB[kTH];
    acc = wmma4(a, b, acc);
    pB += (size_t)4 * kTH;
  }
  const float bv = bi1[n0 + l16];
#pragma unroll
  for (int r = 0; r < 8; ++r)
    ih[(size_t)(t * 16 + r + kh * 8) * kTH + n0 + l16] = fmaxf(acc[r] + bv, 0.0f);
}

// ============================================================
// Interaction GEMM2 + off-diagonal reduce:
//   interact[t][n] = sum_{j != t} (ih[t*16+j] @ Wi2)[n] + 15*bi2[n]
// ============================================================
__global__ void k_inter2(const float* __restrict__ ih,
                         const float* __restrict__ Wi2,
                         const float* __restrict__ bi2,
                         float* __restrict__ interact) {
  const int nt   = blockIdx.x;   // 0..15
  const int t    = blockIdx.y;   // 0..15
  const int lane = threadIdx.x & 31;
  const int kh   = lane >> 4;
  const int l16  = lane & 15;
  const int n0   = nt * 16;

  v8f acc = {};
  const float* pA = ih + (size_t)(t * 16 + l16) * kTH + kh * 2;
  const float* pB = Wi2 + (size_t)(kh * 2) * kD + n0 + l16;
  for (int k0 = 0; k0 < kTH; k0 += 4) {
    v2f a = *(const v2f*)pA;
    v2f b;
    b.x = pB[0];
    b.y = pB[kD];
    acc = wmma4(a, b, acc);
    pA += 4;
    pB += (size_t)4 * kD;
  }
  // reduce over rows j != t (row j = r + kh*8 in this lane)
  float s = 0.0f;
#pragma unroll
  for (int r = 0; r < 8; ++r) {
    const int j = r + kh * 8;
    if (j != t) s += acc[r];
  }
  s += __shfl_xor(s, 16, 32);
  if (lane < 16)
    interact[t * kD + n0 + lane] = s + 15.0f * bi2[n0 + lane];
}

// fin = obj + trans + interact  [16,256]
__global__ void k_fin(const float* __restrict__ obj,
                      const float* __restrict__ trans,
                      const float* __restrict__ interact,
                      float* __restrict__ finv) {
  const int idx = blockIdx.x * blockDim.x + threadIdx.x;  // 0..4095
  finv[idx] = obj[idx] + trans[idx] + interact[idx];
}

// ============================================================
// Fused double render: ONE pass over Wr2[1024,65536] computing both
// sigmoid(h_ini @ Wr2 + br2) and sigmoid(h_fin @ Wr2 + br2), then the
// per-pixel sum_o rgb*mask reduction entirely in registers.
// One wave per 16-pixel tile; its 4 column tiles are {c*HW+p : c=0..3}.
// ============================================================
__global__ void k_render2(const float* __restrict__ h_ini,
                          const float* __restrict__ h_fin,
                          const float* __restrict__ Wr2,
                          const float* __restrict__ br2,
                          float* __restrict__ out) {
  const int p0   = blockIdx.x * 16;     // pixel base (0..16368)
  const int lane = threadIdx.x & 31;
  const int kh   = lane >> 4;
  const int l16  = lane & 15;
  const int ldb  = 4 * kHW;             // 65536

  v8f acc[2][4] = {};
  const float* pA0 = h_ini + l16 * kRH + kh * 2;
  const float* pA1 = h_fin + l16 * kRH + kh * 2;
  const float* pB[4];
#pragma unroll
  for (int c = 0; c < 4; ++c)
    pB[c] = Wr2 + (size_t)(kh * 2) * ldb + c * kHW + p0 + l16;

  for (int k0 = 0; k0 < kRH; k0 += 4) {
    v2f a0 = *(const v2f*)pA0;
    v2f a1 = *(const v2f*)pA1;
#pragma unroll
    for (int c = 0; c < 4; ++c) {
      v2f b;
      b.x = pB[c][0];
      b.y = pB[c][ldb];
      __builtin_prefetch((const void*)(pB[c] + (size_t)16 * ldb), 0, 0);
      acc[0][c] = wmma4(a0, b, acc[0][c]);
      acc[1][c] = wmma4(a1, b, acc[1][c]);
      pB[c] += (size_t)4 * ldb;
    }
    pA0 += 4;
    pA1 += 4;
  }

  // bias + sigmoid (column bias is per-lane scalar, same for all 8 rows)
  float bc[4];
#pragma unroll
  for (int c = 0; c < 4; ++c) bc[c] = br2[c * kHW + p0 + l16];
#pragma unroll
  for (int h = 0; h < 2; ++h)
#pragma unroll
    for (int c = 0; c < 4; ++c)
#pragma unroll
      for (int r = 0; r < 8; ++r)
        acc[h][c][r] = sigmoidf_(acc[h][c][r] + bc[c]);

  // recon[c][p] = sum_o rgb[o][c][p] * mask[o][p]; rows = objects
#pragma unroll
  for (int h = 0; h < 2; ++h) {
#pragma unroll
    for (int c = 0; c < 3; ++c) {
      float s = 0.0f;
#pragma unroll
      for (int r = 0; r < 8; ++r) s += acc[h][c][r] * acc[h][3][r];
      s += __shfl_xor(s, 16, 32);   // combine objects 0-7 with 8-15
      if (lane < 16)
        out[h * 3 * kHW + c * kHW + p0 + lane] = s;
    }
  }
}

// ============================================================
extern "C" void kernel_launch(void* const* d_in, const int* in_sizes, int n_in,
                              void* d_out, int out_size, void* d_ws, size_t ws_size,
                              hipStream_t stream) {
  (void)in_sizes; (void)n_in; (void)out_size; (void)ws_size;
  const float* img   = (const float*)d_in[0];
  const float* masks = (const float*)d_in[1];
  const float* Wp1   = (const float*)d_in[3];
  const float* bp1   = (const float*)d_in[4];
  const float* Wp2   = (const float*)d_in[5];
  const float* bp2   = (const float*)d_in[6];
  const float* Wr1   = (const float*)d_in[7];
  const float* br1   = (const float*)d_in[8];
  const float* Wr2   = (const float*)d_in[9];
  const float* br2   = (const float*)d_in[10];
  const float* Wt1   = (const float*)d_in[11];
  const float* bt1   = (const float*)d_in[12];
  const float* Wt2   = (const float*)d_in[13];
  const float* bt2   = (const float*)d_in[14];
  const float* Wi1   = (const float*)d_in[15];
  const float* bi1   = (const float*)d_in[16];
  const float* Wi2   = (const float*)d_in[17];
  const float* bi2   = (const float*)d_in[18];

  float* ws       = (float*)d_ws;
  float* Ppart    = ws;                    // 16*16*1024 = 262144
  float* C1       = Ppart + 262144;        // 16*1024    = 16384
  float* obj      = C1 + 16384;            // 16*256     = 4096
  float* ht       = obj + 4096;            // 16*512     = 8192
  float* trans    = ht + 8192;             // 16*256     = 4096
  float* ih       = trans + 4096;          // 256*512    = 131072
  float* interact = ih + 131072;           // 16*256     = 4096
  float* finv     = interact + 4096;       // 16*256     = 4096
  float* hini     = finv + 4096;           // 16*1024    = 16384
  float* hfin     = hini + 16384;          // 16*1024    = 16384

  // perception: pin @ Wp1 (split-K over 1024 waves), reduce, @ Wp2
  k_percept1<<<dim3(64, 16), 32, 0, stream>>>(img, masks, Wp1, Ppart);
  k_reduce1<<<64, 256, 0, stream>>>(Ppart, bp1, C1);
  k_gemm16<<<16, 32, 0, stream>>>(C1, kPH, Wp2, kD, bp2, obj, kD, kPH, 0);

  // physics transition
  k_gemm16<<<32, 32, 0, stream>>>(obj, kD, Wt1, kTH, bt1, ht, kTH, kD, 1);
  k_gemm16<<<16, 32, 0, stream>>>(ht, kTH, Wt2, kD, bt2, trans, kD, kTH, 0);

  // pairwise interaction + off-diagonal reduce
  k_inter1<<<dim3(32, 16), 32, 0, stream>>>(obj, Wi1, bi1, ih);
  k_inter2<<<dim3(16, 16), 32, 0, stream>>>(ih, Wi2, bi2, interact);
  k_fin<<<16, 256, 0, stream>>>(obj, trans, interact, finv);

  // render hidden for both latents, then one fused pass over Wr2
  k_gemm16<<<64, 32, 0, stream>>>(obj, kD, Wr1, kRH, br1, hini, kRH, kD, 1);
  k_gemm16<<<64, 32, 0, stream>>>(finv, kD, Wr1, kRH, br1, hfin, kRH, kD, 1);
  k_render2<<<1024, 32, 0, stream>>>(hini, hfin, Wr2, br2, (float*)d_out);
}